// TransLstm_13821204758620
// MI455X (gfx1250) — compile-verified
//
#include <hip/hip_runtime.h>
#include <hip/hip_bf16.h>
#include <math.h>
#include <stdint.h>

// ---------------- problem constants ----------------
#define BATCH 32
#define SEQL  48
#define VOC   16000
#define HID   512
#define NLAY  4
#define G4H   2048      // 4*HID
#define X2H   1024      // 2*HID

typedef __attribute__((ext_vector_type(2))) float v2f;
typedef __attribute__((ext_vector_type(8))) float v8f;

// ------------------------------------------------------------------
// Register-path inner K-loop with one-iteration software pipelining:
// fragments for k+4 are loaded while the 4 WMMAs for k execute, so the
// s_wait_loadcnt can overlap WMMA latency instead of serializing.
// Fragment layout per CDNA5 ISA 7.12.2 (32-bit 16x4 A):
//   lane<16 : row = lane,    holds K+0,K+1 ; lane>=16: row = lane-16, K+2,K+3
// ------------------------------------------------------------------
__device__ __forceinline__ void wmma_k_loop(const float* __restrict__ Az,
                                            const float* __restrict__ Wz,
                                            int m0, int n0, int K,
                                            int lr, int kh, v8f acc[4]) {
  const float* ap = Az + (size_t)(m0 + lr) * K + kh;
  const float* wp = Wz + (size_t)(n0 + lr) * K + kh;
  v2f a_c = *(const v2f*)ap;
  v2f b_c[4];
#pragma unroll
  for (int nt = 0; nt < 4; ++nt)
    b_c[nt] = *(const v2f*)(wp + (size_t)nt * 16 * K);
  for (int k = 0; k < K; k += 4) {
    v2f a_n = a_c;
    v2f b_n[4];
#pragma unroll
    for (int nt = 0; nt < 4; ++nt) b_n[nt] = b_c[nt];
    if (k + 4 < K) {                      // prefetch next iteration
      a_n = *(const v2f*)(ap + k + 4);
#pragma unroll
      for (int nt = 0; nt < 4; ++nt)
        b_n[nt] = *(const v2f*)(wp + (size_t)nt * 16 * K + k + 4);
    }
#pragma unroll
    for (int nt = 0; nt < 4; ++nt)
      acc[nt] = __builtin_amdgcn_wmma_f32_16x16x4_f32(
          false, a_c, false, b_c[nt], (short)0, acc[nt], false, false);
    a_c = a_n;
#pragma unroll
    for (int nt = 0; nt < 4; ++nt) b_c[nt] = b_n[nt];
  }
}

// Generic GEMM (register path): Cout[z] = (Cin[z] | bias[z] | 0) + A[z] @ W[z]^T
// grid: (N/64, M/16, Z), block: 32 (one wave -> EXEC all ones for WMMA)
__global__ __launch_bounds__(32) void wmma_gemm_f32(
    const float* __restrict__ A, const float* __restrict__ W,
    const float* __restrict__ bias, const float* __restrict__ Cin,
    float* __restrict__ Cout, int K, int ldc,
    long long aZ, long long wZ, long long bZ, long long cZ) {
  const int z    = blockIdx.z;
  const int lane = threadIdx.x;
  const int lr   = lane & 15;
  const int kh   = (lane >> 4) << 1;
  const int rh   = (lane >> 4) << 3;
  const int m0   = blockIdx.y << 4;
  const int n0   = blockIdx.x << 6;
  const float* Az = A + (size_t)z * aZ;
  const float* Wz = W + (size_t)z * wZ;

  v8f acc[4] = {};
  if (Cin) {
    const float* Cz = Cin + (size_t)z * cZ;
#pragma unroll
    for (int nt = 0; nt < 4; ++nt) {
      int col = n0 + nt * 16 + lr;
#pragma unroll
      for (int v = 0; v < 8; ++v)
        acc[nt][v] = Cz[(size_t)(m0 + v + rh) * ldc + col];
    }
  } else if (bias) {
    const float* Bz = bias + (size_t)z * bZ;
#pragma unroll
    for (int nt = 0; nt < 4; ++nt) {
      float bv = Bz[n0 + nt * 16 + lr];
#pragma unroll
      for (int v = 0; v < 8; ++v) acc[nt][v] = bv;
    }
  }

  wmma_k_loop(Az, Wz, m0, n0, K, lr, kh, acc);

  float* Co = Cout + (size_t)z * cZ;
#pragma unroll
  for (int nt = 0; nt < 4; ++nt) {
    int col = n0 + nt * 16 + lr;
#pragma unroll
    for (int v = 0; v < 8; ++v)
      Co[(size_t)(m0 + v + rh) * ldc + col] = acc[nt][v];
  }
}

// ------------------------------------------------------------------
// LDS-staged GEMM for the bandwidth-heavy batched GEMMs.
// NW waves per block share one 64(N) x 16(K) W tile staged in LDS by the
// CDNA5 async-to-LDS DMA (global_load_async_to_lds_b128, ASYNCcnt),
// double-buffered so chunk k+1 streams in while chunk k is consumed.
// Each wave owns a different 16-row m-tile -> W traffic divided by NW.
// LDS rows padded to 18 floats: ds_load_b64 lane stride = 18 banks, no
// conflicts; all offsets stay 8B aligned.
// grid: (N/64, M/(16*NW), Z), block: NW*32
// ------------------------------------------------------------------
template <int NW>
__global__ __launch_bounds__(NW * 32) void wmma_gemm_lds(
    const float* __restrict__ A, const float* __restrict__ W,
    const float* __restrict__ bias, float* __restrict__ Cout,
    int K, int ldc, long long aZ, long long wZ, long long bZ, long long cZ) {
  constexpr int PER = 256 / (NW * 32);   // async b128 issues per thread/chunk
  __shared__ __align__(16) float ldsW[2][64][18];
  const int z    = blockIdx.z;
  const int tid  = threadIdx.x;
  const int w    = tid >> 5;
  const int lane = tid & 31;
  const int lr   = lane & 15;
  const int kh   = (lane >> 4) << 1;
  const int rh   = (lane >> 4) << 3;
  const int m0   = (blockIdx.y * NW + w) << 4;
  const int n0   = blockIdx.x << 6;
  const float* Az = A + (size_t)z * aZ;
  const float* Wz = W + (size_t)z * wZ;

  v8f acc[4] = {};
  if (bias) {
    const float* Bz = bias + (size_t)z * bZ;
#pragma unroll
    for (int nt = 0; nt < 4; ++nt) {
      float bv = Bz[n0 + nt * 16 + lr];
#pragma unroll
      for (int v = 0; v < 8; ++v) acc[nt][v] = bv;
    }
  }

  // async-stage one 64x16 W chunk (64 rows x 64B = 256 b128 transfers)
  auto stage = [&](int b, int kc) {
#pragma unroll
    for (int j = 0; j < PER; ++j) {
      int id = tid + j * (NW * 32);
      int r = id >> 2, q = (id & 3) << 2;
      unsigned la = (unsigned)(uintptr_t)&ldsW[b][r][q];
      unsigned long long ga =
          (unsigned long long)(uintptr_t)(Wz + (size_t)(n0 + r) * K + kc + q);
      asm volatile("global_load_async_to_lds_b128 %0, %1, off"
                   :: "v"(la), "v"(ga) : "memory");
    }
  };

  stage(0, 0);
  int buf = 0;
  for (int kc = 0; kc < K; kc += 16) {
    if (kc + 16 < K) {
      stage(buf ^ 1, kc + 16);                 // overlap DMA with compute
      asm volatile("s_wait_asynccnt %0" :: "i"(PER) : "memory");
    } else {
      asm volatile("s_wait_asynccnt 0" ::: "memory");
    }
    __syncthreads();                            // chunk `buf` visible to all
    v2f af[4];
#pragma unroll
    for (int j = 0; j < 4; ++j)
      af[j] = *(const v2f*)(Az + (size_t)(m0 + lr) * K + kc + j * 4 + kh);
#pragma unroll
    for (int j = 0; j < 4; ++j) {
#pragma unroll
      for (int nt = 0; nt < 4; ++nt) {
        v2f bf = *(const v2f*)&ldsW[buf][nt * 16 + lr][j * 4 + kh];
        acc[nt] = __builtin_amdgcn_wmma_f32_16x16x4_f32(
            false, af[j], false, bf, (short)0, acc[nt], false, false);
      }
    }
    __syncthreads();                            // reads done before overwrite
    buf ^= 1;
  }

  float* Co = Cout + (size_t)z * cZ;
#pragma unroll
  for (int nt = 0; nt < 4; ++nt) {
    int col = n0 + nt * 16 + lr;
#pragma unroll
    for (int v = 0; v < 8; ++v)
      Co[(size_t)(m0 + v + rh) * ldc + col] = acc[nt][v];
  }
}

// Encoder recurrent step: GX[dir][t_dir] += h[dir] @ Whh[dir]^T (in place).
__global__ __launch_bounds__(32) void enc_rec_gemm(
    const float* __restrict__ h, const float* __restrict__ Whh,
    float* __restrict__ GX, int tf, int tb) {
  const int z  = blockIdx.z;
  const int tz = z ? tb : tf;
  float* G = GX + (size_t)z * (SEQL * BATCH * G4H) + (size_t)tz * (BATCH * G4H);
  const float* Az = h   + (size_t)z * (BATCH * HID);
  const float* Wz = Whh + (size_t)z * (G4H * HID);
  const int lane = threadIdx.x;
  const int lr = lane & 15, kh = (lane >> 4) << 1, rh = (lane >> 4) << 3;
  const int m0 = blockIdx.y << 4, n0 = blockIdx.x << 6;

  v8f acc[4];
#pragma unroll
  for (int nt = 0; nt < 4; ++nt) {
    int col = n0 + nt * 16 + lr;
#pragma unroll
    for (int v = 0; v < 8; ++v)
      acc[nt][v] = G[(size_t)(m0 + v + rh) * G4H + col];
  }
  wmma_k_loop(Az, Wz, m0, n0, HID, lr, kh, acc);
#pragma unroll
  for (int nt = 0; nt < 4; ++nt) {
    int col = n0 + nt * 16 + lr;
#pragma unroll
    for (int v = 0; v < 8; ++v)
      G[(size_t)(m0 + v + rh) * G4H + col] = acc[nt][v];
  }
}

// Fused LSTM pointwise: gates -> (h,c) update, h written to concat output.
__global__ void lstm_pw(const float* __restrict__ G, long long g0, long long g1,
                        float* __restrict__ h, float* __restrict__ c, long long hcZ,
                        float* __restrict__ out, long long o0, long long o1) {
  const int z = blockIdx.z;
  const float* g = G + (z ? g1 : g0);
  int i = blockIdx.x * blockDim.x + threadIdx.x;  // [0, B*H)
  int b = i >> 9, n = i & (HID - 1);
  const float* row = g + (size_t)b * G4H;
  float ig = 1.f / (1.f + expf(-row[n]));
  float fg = 1.f / (1.f + expf(-row[HID + n]));
  float gg = tanhf(row[2 * HID + n]);
  float og = 1.f / (1.f + expf(-row[3 * HID + n]));
  size_t hi = (size_t)z * hcZ + (size_t)b * HID + n;
  float cn = fg * c[hi] + ig * gg;
  c[hi] = cn;
  float hn = og * tanhf(cn);
  h[hi] = hn;
  out[(z ? o1 : o0) + (size_t)b * X2H + n] = hn;
}

// Embedding lookup, time-major: x0[t][b][:] = emb[data[b][t]]
__global__ void embed_k(const float* __restrict__ emb, const int* __restrict__ data,
                        float* __restrict__ x0) {
  int t = blockIdx.x >> 5, b = blockIdx.x & 31;
  int tok = data[b * SEQL + t];
  const float* src = emb + (size_t)tok * HID;
  float* dst = x0 + (size_t)blockIdx.x * HID;
  for (int hh = threadIdx.x; hh < HID; hh += blockDim.x) dst[hh] = src[hh];
}

// SOS one-hot row at t=0 for every batch element.
__global__ void sos_k(float* __restrict__ out, const int* __restrict__ data) {
  int tok = data[0];                       // data[0,0]
  int i = blockIdx.x * 256 + threadIdx.x;  // [0, B*V)
  int b = i / VOC, v = i - b * VOC;
  out[(size_t)b * SEQL * VOC + v] = (v == tok) ? 1.0f : 0.0f;
}

// Row argmax with first-index tie-break (matches jnp.argmax). grid=B, block=256.
__global__ void argmax_k(const float* __restrict__ logits, int* __restrict__ idx) {
  __shared__ float sv[256];
  __shared__ int si[256];
  const float* row = logits + (size_t)blockIdx.x * (SEQL * VOC);
  float best = -3.402823466e+38f;
  int bi = 0;
  for (int v = threadIdx.x; v < VOC; v += 256) {
    float x = row[v];
    if (x > best) { best = x; bi = v; }
  }
  sv[threadIdx.x] = best; si[threadIdx.x] = bi;
  __syncthreads();
  for (int s = 128; s > 0; s >>= 1) {
    if (threadIdx.x < s) {
      float ov = sv[threadIdx.x + s]; int oi = si[threadIdx.x + s];
      if (ov > sv[threadIdx.x] || (ov == sv[threadIdx.x] && oi < si[threadIdx.x])) {
        sv[threadIdx.x] = ov; si[threadIdx.x] = oi;
      }
    }
    __syncthreads();
  }
  if (threadIdx.x == 0) idx[blockIdx.x] = si[0];
}

// Feedback: x[b][:] = emb[idx[b]] (512-wide, packed). grid=B, block=256.
__global__ void gather_k(const float* __restrict__ emb, const int* __restrict__ idx,
                         float* __restrict__ x) {
  const float* src = emb + (size_t)idx[blockIdx.x] * HID;
  float* dst = x + (size_t)blockIdx.x * HID;
  for (int hh = threadIdx.x; hh < HID; hh += 256) dst[hh] = src[hh];
}

extern "C" void kernel_launch(void* const* d_in, const int* in_sizes, int n_in,
                              void* d_out, int out_size, void* d_ws, size_t ws_size,
                              hipStream_t stream) {
  const int*   data  = (const int*)d_in[0];
  const float* emb   = (const float*)d_in[1];
  const float* eWih0 = (const float*)d_in[2];
  const float* eWih  = (const float*)d_in[3];
  const float* eWhh  = (const float*)d_in[4];
  const float* eb    = (const float*)d_in[5];
  const float* dWih0 = (const float*)d_in[6];
  const float* dWih  = (const float*)d_in[7];
  const float* dWhh  = (const float*)d_in[8];
  const float* db    = (const float*)d_in[9];
  const float* Wout  = (const float*)d_in[10];
  float* out = (float*)d_out;

  // ---- workspace layout (~40 MB of floats) ----
  float* XA  = (float*)d_ws;                        // L*B*2H xs ping
  float* XB  = XA + (size_t)SEQL * BATCH * X2H;     // L*B*2H xs pong
  float* GX  = XB + (size_t)SEQL * BATCH * X2H;     // 2 * L*B*4H gate pre-acts
  float* HS  = GX + (size_t)2 * SEQL * BATCH * G4H; // NLAY*2*B*H hidden
  float* CS  = HS + (size_t)NLAY * 2 * BATCH * HID; // NLAY*2*B*H cell
  float* DG  = CS + (size_t)NLAY * 2 * BATCH * HID; // 2*B*4H decoder gates
  float* DXA = DG + (size_t)2 * BATCH * G4H;        // B*2H decoder x ping
  float* DXB = DXA + (size_t)BATCH * X2H;           // B*2H decoder x pong
  int*   IDX = (int*)(DXB + (size_t)BATCH * X2H);   // B argmax indices

  const long long LB4H = (long long)SEQL * BATCH * G4H;
  const long long B4H  = (long long)BATCH * G4H;
  const long long BH   = (long long)BATCH * HID;

  embed_k<<<dim3(SEQL * BATCH), dim3(128), 0, stream>>>(emb, data, XA);
  hipMemcpyAsync(DXA, XA, (size_t)BATCH * HID * sizeof(float),
                 hipMemcpyDeviceToDevice, stream);
  hipMemsetAsync(HS, 0, (size_t)NLAY * 2 * BATCH * HID * sizeof(float), stream);
  hipMemsetAsync(CS, 0, (size_t)NLAY * 2 * BATCH * HID * sizeof(float), stream);
  sos_k<<<dim3((BATCH * VOC) / 256), dim3(256), 0, stream>>>(out, data);

  // ---------------- encoder ----------------
  float* xin = XA;
  float* xout = XB;
  for (int l = 0; l < NLAY; ++l) {
    int K = l ? X2H : HID;
    const float* Wih  = l ? (eWih + (size_t)(l - 1) * 2 * G4H * X2H) : eWih0;
    const float* bias = eb + (size_t)l * 2 * G4H;
    // time-batched input GEMM (async-LDS staged, 4 waves share W tiles)
    wmma_gemm_lds<4><<<dim3(G4H / 64, (SEQL * BATCH) / 64, 2), 128, 0, stream>>>(
        xin, Wih, bias, GX, K, G4H,
        0LL, (long long)G4H * K, (long long)G4H, LB4H);
    const float* Whh = eWhh + (size_t)l * 2 * G4H * HID;
    float* h = HS + (size_t)l * 2 * BATCH * HID;
    float* c = CS + (size_t)l * 2 * BATCH * HID;
    for (int s = 0; s < SEQL; ++s) {
      int tf = s, tb = SEQL - 1 - s;
      enc_rec_gemm<<<dim3(G4H / 64, 2, 2), 32, 0, stream>>>(h, Whh, GX, tf, tb);
      lstm_pw<<<dim3((BATCH * HID) / 256, 1, 2), 256, 0, stream>>>(
          GX, (long long)tf * B4H, LB4H + (long long)tb * B4H,
          h, c, BH,
          xout, (long long)tf * BATCH * X2H, (long long)tb * BATCH * X2H + HID);
    }
    float* t = xin; xin = xout; xout = t;
  }

  // ---------------- decoder (greedy) ----------------
  for (int t = 0; t < SEQL - 1; ++t) {
    float* dx = DXA;
    float* dxo = DXB;
    for (int l = 0; l < NLAY; ++l) {
      int Kx = l ? X2H : HID;
      const float* Wih  = l ? (dWih + (size_t)(l - 1) * 2 * G4H * X2H) : dWih0;
      const float* bias = db + (size_t)l * 2 * G4H;
      // DG[dir] = x @ Wih[dir]^T + b[dir]
      wmma_gemm_f32<<<dim3(G4H / 64, BATCH / 16, 2), 32, 0, stream>>>(
          dx, Wih, bias, nullptr, DG, Kx, G4H,
          0LL, (long long)G4H * Kx, (long long)G4H, B4H);
      const float* Whh = dWhh + (size_t)l * 2 * G4H * HID;
      float* h = HS + (size_t)l * 2 * BATCH * HID;
      float* c = CS + (size_t)l * 2 * BATCH * HID;
      // DG[dir] += h[dir] @ Whh[dir]^T   (in-place accumulate)
      wmma_gemm_f32<<<dim3(G4H / 64, BATCH / 16, 2), 32, 0, stream>>>(
          h, Whh, nullptr, DG, DG, HID, G4H,
          BH, (long long)G4H * HID, 0LL, B4H);
      lstm_pw<<<dim3((BATCH * HID) / 256, 1, 2), 256, 0, stream>>>(
          DG, 0LL, B4H, h, c, BH, dxo, 0LL, (long long)HID);
      float* tp = dx; dx = dxo; dxo = tp;
    }
    // dx == DXA holds the final 2H concat; logits -> d_out[:, t+1, :]
    float* lrow = out + (size_t)(t + 1) * VOC;
    wmma_gemm_lds<2><<<dim3(VOC / 64, 1, 1), 64, 0, stream>>>(
        dx, Wout, nullptr, lrow, X2H, SEQL * VOC, 0LL, 0LL, 0LL, 0LL);
    argmax_k<<<dim3(BATCH), dim3(256), 0, stream>>>(lrow, IDX);
    gather_k<<<dim3(BATCH), dim3(256), 0, stream>>>(emb, IDX, DXA);
  }
}